// GraphNN_KNN_v1_v0_17970143167396
// MI455X (gfx1250) — compile-verified
//
#include <hip/hip_runtime.h>

#define NN      100000            // N_NODES
#define NE      6400000           // N_EDGES
#define NORD    4
#define EPO     (NE / NORD)       // 1,600,000
#define KF      10                // feature width K
#define K2      (2 * KF)          // 20
#define HPAD    16                // padded row stride for h (64B aligned rows)
#define OUTPAD  12                // staged row stride in k_out

typedef __attribute__((ext_vector_type(2))) float v2f;
typedef __attribute__((ext_vector_type(8))) float v8f;

// ---------------------------------------------------------------------------
// h0[node][c] = (c < K) ? x[node][c] : 0      (pad columns stay zero forever)
// ---------------------------------------------------------------------------
__global__ void k_init(const float* __restrict__ x, float* __restrict__ h) {
    int i = blockIdx.x * blockDim.x + threadIdx.x;
    if (i >= NN * HPAD) return;
    int node = i / HPAD, c = i % HPAD;
    h[i] = (c < KF) ? x[node * KF + c] : 0.0f;
}

// ---------------------------------------------------------------------------
// One message-passing step. 1 wave = 16 edges.
//  phase 1 (lanes 0-15, half EXEC): edge lookup, gather x_i/x_j, build the
//          20-wide msg row, stage it in LDS (5x ds_store_b128).
//  phase 2 (full EXEC): ds_load_b64 A-fragments at the lane-varying address
//          (m*20 + 4c + 2*hi), 5 accumulating v_wmma_f32_16x16x4_f32.
//  phase 3: 8 full-EXEC shuffles fetch the dst row ids, then ONE exec-masked
//          region does all 8 scatter atomics with 32-bit saddr+voffset form.
// D columns n >= 10 are dead (never stored) so B needs no zero-masking.
// ---------------------------------------------------------------------------
__global__ void __launch_bounds__(256)
k_edges(const float* __restrict__ hcur, float* __restrict__ hnext,
        const float* __restrict__ w_mp, const float* __restrict__ b_mp,
        const int* __restrict__ edge_index, const int* __restrict__ order) {
    __shared__ float stage[8][16 * K2];              // 10,240 B, per-wave tiles

    const int lane = threadIdx.x & 31;
    const int wave = threadIdx.x >> 5;
    const int tile = blockIdx.x * 8 + wave;          // 100,000 tiles total
    const int e0   = tile * 16;                      // < 1.6M: 32-bit safe

    const int  m  = lane & 15;                       // edge slot / C-D column
    const bool hi = (lane >> 4) != 0;                // which K-pair this lane holds
    const int  n  = m;
    const int  nc = (n < KF) ? n : 0;                // clamped (in-bounds) column

    float* ws = &stage[wave][0];

    int dst = 0;
    if (lane < 16) {                                 // gather phase: half EXEC
        int o   = order[e0 + m];
        dst     = edge_index[o];
        int src = edge_index[NE + o];

        // 32-bit element offsets (max 1.6M) -> saddr + voffset addressing
        const float4* pi = (const float4*)(hcur + dst * HPAD);
        const float4* pj = (const float4*)(hcur + src * HPAD);
        float4 i0 = pi[0], i1 = pi[1], i2 = pi[2];
        float4 j0 = pj[0], j1 = pj[1], j2 = pj[2];
        float xi[KF] = { i0.x,i0.y,i0.z,i0.w, i1.x,i1.y,i1.z,i1.w, i2.x,i2.y };
        float xj[KF] = { j0.x,j0.y,j0.z,j0.w, j1.x,j1.y,j1.z,j1.w, j2.x,j2.y };

        float msg[K2];                               // constant-indexed -> SROA
#pragma unroll
        for (int k = 0; k < KF; ++k) { msg[k] = xi[k]; msg[KF+k] = xj[k]-xi[k]; }

        float4* sp = (float4*)(ws + m * K2);         // 80m bytes: 16B aligned
#pragma unroll
        for (int q = 0; q < 5; ++q)
            sp[q] = make_float4(msg[4*q], msg[4*q+1], msg[4*q+2], msg[4*q+3]);
    }
    __builtin_amdgcn_wave_barrier();                 // order DS store -> DS load

    // accumulator seeded with per-column bias (dead columns may hold junk bias)
    float cinit = b_mp[nc];
    v8f acc;
#pragma unroll
    for (int j = 0; j < 8; ++j) acc[j] = cinit;

    // B rows for this lane: one unconditional b64 gather per chunk.
    // address (nc*20 + 4c + 2*hi) is even and <= 199 -> in-bounds, 8B aligned.
    const float2* wrow = (const float2*)(w_mp + nc * K2);
    const float*  arow = ws + m * K2 + (hi ? 2 : 0);

    // K = 20 -> 5 chunks of 4
#pragma unroll
    for (int c = 0; c < 5; ++c) {
        float2 av = *(const float2*)(arow + 4 * c); // ds_load_b64, dynamic addr
        float2 bv = wrow[2 * c + (hi ? 1 : 0)];     // global_load_b64 gather
        v2f a, b;
        a.x = av.x; a.y = av.y;
        b.x = bv.x; b.y = bv.y;
        acc = __builtin_amdgcn_wmma_f32_16x16x4_f32(
                  false, a, false, b, (short)0, acc, false, false);
    }

    // Fetch all 8 dst row ids with FULL EXEC (bpermute reads lanes 0-15;
    // disabling lanes 10-15 before the shuffles would corrupt them).
    int off[8];
#pragma unroll
    for (int j = 0; j < 8; ++j) {
        int row = j + (hi ? 8 : 0);
        int d   = __shfl(dst, row, 32);
        off[j]  = d * HPAD + n;                      // 32-bit element offset
    }

    // One exec-masked region for all 8 scatter atomics.
    if (n < KF) {
#pragma unroll
        for (int j = 0; j < 8; ++j)
            unsafeAtomicAdd(hnext + off[j], acc[j]);
    }
}

// ---------------------------------------------------------------------------
// out = h @ w_out^T + b_out, 1 wave = 16 nodes, 3x wmma (K 10 padded to 12;
// h pad columns are zero, and B addresses are clamped so dead K rows multiply
// against zero A elements). Same LDS staging for the A fragments.
// ---------------------------------------------------------------------------
__global__ void __launch_bounds__(256)
k_out(const float* __restrict__ h, const float* __restrict__ w_out,
      const float* __restrict__ b_out, float* __restrict__ out) {
    __shared__ float stage[8][16 * OUTPAD];          // 6,144 B

    const int lane = threadIdx.x & 31;
    const int wave = threadIdx.x >> 5;
    const int tile = blockIdx.x * 8 + wave;
    if (tile * 16 >= NN) return;                     // whole-wave exit: EXEC stays full
    const int nodeBase = tile * 16;

    const int  m  = lane & 15;
    const bool hi = (lane >> 4) != 0;
    const int  n  = m;
    const int  nc = (n < KF) ? n : 0;

    float* ws = &stage[wave][0];

    if (lane < 16) {                                 // stage 16 h rows (12 floats)
        const float4* ph = (const float4*)(h + (nodeBase + m) * HPAD);
        float4 h0 = ph[0], h1 = ph[1], h2 = ph[2];   // cols 10,11 are zero pad
        float4* sp = (float4*)(ws + m * OUTPAD);     // 48m bytes: 16B aligned
        sp[0] = h0; sp[1] = h1; sp[2] = h2;
    }
    __builtin_amdgcn_wave_barrier();

    float cinit = b_out[nc];
    v8f acc;
#pragma unroll
    for (int j = 0; j < 8; ++j) acc[j] = cinit;

    const float* arow = ws + m * OUTPAD + (hi ? 2 : 0);

#pragma unroll
    for (int c = 0; c < 3; ++c) {
        const int k0 = 4 * c + (hi ? 2 : 0);
        const int kb = (k0 < KF) ? k0 : 0;           // clamp: dead K rows hit A==0
        float2 av = *(const float2*)(arow + 4 * c);  // ds_load_b64
        float2 bv = *(const float2*)(w_out + nc * KF + kb); // even addr: aligned
        v2f a, b;
        a.x = av.x; a.y = av.y;
        b.x = bv.x; b.y = bv.y;
        acc = __builtin_amdgcn_wmma_f32_16x16x4_f32(
                  false, a, false, b, (short)0, acc, false, false);
    }

    // One exec-masked region for all 8 stores, 32-bit offsets.
    if (n < KF) {
        const int base = (nodeBase + (hi ? 8 : 0)) * KF + n;
#pragma unroll
        for (int j = 0; j < 8; ++j)
            out[base + j * KF] = acc[j];
    }
}

// ---------------------------------------------------------------------------
extern "C" void kernel_launch(void* const* d_in, const int* in_sizes, int n_in,
                              void* d_out, int out_size, void* d_ws, size_t ws_size,
                              hipStream_t stream) {
    const float* x      = (const float*)d_in[0];
    const float* w_mp   = (const float*)d_in[1];
    const float* b_mp   = (const float*)d_in[2];
    const float* w_out  = (const float*)d_in[3];
    const float* b_out  = (const float*)d_in[4];
    const int*   eidx   = (const int*)d_in[5];
    const int*   orders = (const int*)d_in[6];
    float*       out    = (float*)d_out;

    float* hbuf0 = (float*)d_ws;
    float* hbuf1 = hbuf0 + (size_t)NN * HPAD;
    const size_t hbytes = (size_t)NN * HPAD * sizeof(float);

    {   // pad-init h0 from x
        int tot = NN * HPAD;
        k_init<<<(tot + 255) / 256, 256, 0, stream>>>(x, hbuf0);
    }

    float* cur = hbuf0;
    float* nxt = hbuf1;
    for (int s = 0; s < NORD; ++s) {
        // accumulator starts at h (aggr + h), then edges atomically add msgs
        hipMemcpyAsync(nxt, cur, hbytes, hipMemcpyDeviceToDevice, stream);
        // EPO/128 = 12,500 blocks x 8 waves x 16 edges = 1.6M edges
        k_edges<<<EPO / 128, 256, 0, stream>>>(cur, nxt, w_mp, b_mp,
                                               eidx, orders + (size_t)s * EPO);
        float* t = cur; cur = nxt; nxt = t;
    }

    // 6250 node-tiles of 16, 8 waves/block
    k_out<<<(6250 + 7) / 8, 256, 0, stream>>>(cur, w_out, b_out, out);
}